// MaskedUniformSampler_20547123544753
// MI455X (gfx1250) — compile-verified
//
#include <hip/hip_runtime.h>
#include <stdint.h>

typedef unsigned int v4u __attribute__((ext_vector_type(4)));
typedef int          v8i __attribute__((ext_vector_type(8)));
typedef int          v4i __attribute__((ext_vector_type(4)));
typedef float        v4f __attribute__((ext_vector_type(4)));

#define NUM_ITEMS_C 1000000
#define SORT_N 256

// One block per user row. TDM pulls the 200-entry history into LDS with a
// 256-wide tile (OOB -> zero fill == extra padding zeros). Bitonic sort in
// LDS, branchless rank searches, b128 non-temporal streaming for samples.
__global__ __launch_bounds__(256)
void masked_uniform_sampler_kernel(const float* __restrict__ neg_float,
                                   const int*   __restrict__ user_hist,
                                   float* __restrict__ pos_prob,
                                   int*   __restrict__ neg_out,
                                   float* __restrict__ neg_prob,
                                   int HL, int NUMNEG)
{
    __shared__ int   s_hist[SORT_N];
    __shared__ float s_shift[SORT_N];

    const int tid = threadIdx.x;
    const int b   = blockIdx.x;
    const size_t rowbase = (size_t)b * (size_t)NUMNEG;

    // Warm L2 with this row's samples while the TDM runs.
    __builtin_prefetch(&neg_float[rowbase + (size_t)tid * 4], 0, 3);

    // ---- Tensor Data Mover: DMA history row (global -> LDS) ----
    if (tid < 32) {  // wave 0 issues the descriptor
        uint64_t ga  = (uint64_t)(uintptr_t)(user_hist + (size_t)b * (size_t)HL);
        uint32_t lds = (uint32_t)(uintptr_t)(&s_hist[0]);
        v4u g0;
        g0[0] = 1u;                                              // count=1 (valid D#)
        g0[1] = lds;                                             // lds_addr (bytes)
        g0[2] = (uint32_t)ga;                                    // global_addr[31:0]
        g0[3] = (uint32_t)((ga >> 32) & 0x01FFFFFFu) | 0x80000000u; // addr[56:32], type=2
        v8i g1;
        g1[0] = 0x00020000;                   // data_size = 4 bytes; no multicast/pad
        g1[1] = 0;                            // no atomic barrier
        g1[2] = (int)((HL & 0xFFFF) << 16);   // tensor_dim0 = HL (bits 79:48 low half)
        g1[3] = (int)(SORT_N << 16);          // tile_dim0 = 256 (OOB reads -> 0)
        g1[4] = 0;                            // tile_dim1/2 unused
        g1[5] = HL;                           // tensor_dim0_stride
        g1[6] = 0;
        g1[7] = 0;
        v4i gz4 = {0, 0, 0, 0};               // groups 2/3 unused (<=2D tensor)
        v8i gz8 = {0, 0, 0, 0, 0, 0, 0, 0};   // extra operand on 6-arg toolchain
        __builtin_amdgcn_tensor_load_to_lds(g0, g1, gz4, gz4, gz8, 0);
        __builtin_amdgcn_s_wait_tensorcnt(0);
    }
    __syncthreads();   // LDS tile (200 real + 56 zero-filled) visible to all waves

    // ---- Bitonic sort of 256 ints in LDS (zeros sort to front) ----
    for (int k = 2; k <= SORT_N; k <<= 1) {
        for (int j = k >> 1; j > 0; j >>= 1) {
            int ixj = tid ^ j;
            if (ixj > tid) {
                int a = s_hist[tid];
                int c = s_hist[ixj];
                if (((tid & k) == 0) == (a > c)) {
                    s_hist[tid] = c;
                    s_hist[ixj] = a;
                }
            }
            __syncthreads();
        }
    }

    // pad = #zeros in the sorted 256 array (real padding + 56 OOB zeros).
    // All lanes compute identically -> broadcast LDS reads, no conflicts.
    int pad = 0;
#pragma unroll
    for (int s = 128; s > 0; s >>= 1)
        if (s_hist[pad + s - 1] <= 0) pad += s;

    // shifted[i] = sorted[i] - max(i - pad, 0); matches reference ranks exactly
    // (the 56 extra zeros shift both pos and pad by 56, which cancels).
    int off = tid - pad;
    if (off < 0) off = 0;
    s_shift[tid] = (float)(s_hist[tid] - off);
    __syncthreads();

    // nzc = 256 - pad  ->  scale = NUM_ITEMS - nzc  (exact in f32, < 2^24)
    const float scale = (float)(NUM_ITEMS_C - (SORT_N - pad));
    const float fpad  = (float)pad;
    const v4f zero4 = {0.0f, 0.0f, 0.0f, 0.0f};

    // 4 samples per thread, moved as single b128 non-temporal transactions.
    for (int j = tid * 4; j + 3 < NUMNEG; j += 1024) {
        v4f u4 = __builtin_nontemporal_load((const v4f*)(neg_float + rowbase + j));
        v4i r4;
#pragma unroll
        for (int e = 0; e < 4; ++e) {
            float v = floorf(u4[e] * scale) + 1.0f;
            // branchless upper_bound: pos = #elements <= v
            int pos = 0;
#pragma unroll
            for (int s = 128; s > 0; s >>= 1)
                if (s_shift[pos + s - 1] <= v) pos += s;
            r4[e] = (int)(v + ((float)pos - fpad)); // == v + (rank_ref - padding_ref)
        }
        __builtin_nontemporal_store(r4,   (v4i*)(neg_out  + rowbase + j));
        __builtin_nontemporal_store(zero4,(v4f*)(neg_prob + rowbase + j));
    }
    if (tid == 0) pos_prob[b] = 0.0f;   // -log(1) = 0
}

extern "C" void kernel_launch(void* const* d_in, const int* in_sizes, int n_in,
                              void* d_out, int out_size, void* d_ws, size_t ws_size,
                              hipStream_t stream) {
    (void)n_in; (void)out_size; (void)d_ws; (void)ws_size;
    // inputs: 0=query(f32,unused) 1=neg_float(f32) 2=pos_items(int) 3=user_hist(int) 4=num_neg
    const float* neg_float = (const float*)d_in[1];
    const int*   user_hist = (const int*)d_in[3];
    const int B      = in_sizes[2];            // pos_items count
    const int NUMNEG = in_sizes[1] / B;        // 1024
    const int HL     = in_sizes[3] / B;        // 200

    float* out      = (float*)d_out;
    float* pos_prob = out;                                       // [B] zeros
    int*   neg_out  = (int*)(out + B);                           // [B*NUMNEG] int32
    float* neg_prob = out + (size_t)B + (size_t)B * (size_t)NUMNEG; // [B*NUMNEG] zeros

    masked_uniform_sampler_kernel<<<B, 256, 0, stream>>>(
        neg_float, user_hist, pos_prob, neg_out, neg_prob, HL, NUMNEG);
}